// ProtoSimLoss_intra_calss_65687229826063
// MI455X (gfx1250) — compile-verified
//
#include <hip/hip_runtime.h>
#include <math.h>

// Problem constants (match reference)
#define B_ROWS 16384      // base batch
#define N_ROWS 32768      // V*B rows of cf
#define DIMS   512
#define K_GRP  1024       // 64 subjects * 16 labels
#define NLAB   16
#define APITCH 520        // 512 + 8 pad (floats) to spread LDS banks
#define SPITCH 1040       // 1024 + 16 pad (floats)
#define MROWS  32         // rows per block in the GEMM/loss kernel

typedef __attribute__((ext_vector_type(2))) float v2f;
typedef __attribute__((ext_vector_type(8))) float v8f;

// ---------------- K0: gid = subject*16 + labels ----------------
__global__ void gid_kernel(const int* __restrict__ subject,
                           const int* __restrict__ labels,
                           int* __restrict__ gid) {
  int i = blockIdx.x * blockDim.x + threadIdx.x;
  if (i < B_ROWS) gid[i] = subject[i] * NLAB + labels[i];
}

// ---------------- K1: centroids (+ transpose) + counts (one block per group) ----------------
__global__ void centroid_kernel(const float* __restrict__ X,
                                const int* __restrict__ gid,
                                float* __restrict__ centroid,
                                float* __restrict__ centroidT,
                                float* __restrict__ counts) {
  const int g = blockIdx.x;
  const int t = threadIdx.x; // 256 threads, each owns dims t and 256+t
  float a0 = 0.0f, a1 = 0.0f;
  int match = 0;
  for (int i = 0; i < B_ROWS; ++i) {
    if (gid[i] == g) {
      const float* p = X + (size_t)i * 1024; // [B,V,D] row-major, V*D = 1024
      a0 += p[t]       + p[512 + t];         // view 0 + view 1, dim t
      a1 += p[256 + t] + p[768 + t];         // dim 256+t
      ++match;
    }
  }
  const float cnt = 2.0f * (float)match;
  const float sc  = 1.0f / fmaxf(cnt, 1.0f);
  const float c0 = a0 * sc, c1 = a1 * sc;
  centroid[(size_t)g * DIMS + t]            = c0;
  centroid[(size_t)g * DIMS + 256 + t]      = c1;
  centroidT[(size_t)t * K_GRP + g]          = c0;   // [D][K] for coalesced B loads
  centroidT[(size_t)(256 + t) * K_GRP + g]  = c1;
  if (t == 0) counts[g] = cnt;
}

// ---------------- K2: per-row sqrt(dist) (one wave per row) ----------------
__global__ void dist_kernel(const float* __restrict__ X,
                            const int* __restrict__ gid,
                            const float* __restrict__ centroid,
                            float* __restrict__ sqd) {
  const int wave = threadIdx.x >> 5;
  const int lane = threadIdx.x & 31;
  const int R = blockIdx.x * 8 + wave;
  const int base = (R < B_ROWS) ? R : R - B_ROWS;
  const int view = (R < B_ROWS) ? 0 : 1;
  const float* x = X + (size_t)base * 1024 + view * DIMS;
  const float* c = centroid + (size_t)gid[base] * DIMS;
  float s = 0.0f;
  for (int d = lane; d < DIMS; d += 32) {
    float df = x[d] - c[d];
    s += df * df;
  }
  for (int off = 16; off; off >>= 1) s += __shfl_xor(s, off, 32);
  if (lane == 0) sqd[R] = sqrtf(sqrtf(s)); // sqrt(dist), dist = sqrt(sumsq)
}

// ---------------- K3: per-group raw density ----------------
__global__ void group_density_kernel(const int* __restrict__ gid,
                                     const float* __restrict__ sqd,
                                     const float* __restrict__ counts,
                                     float* __restrict__ density) {
  __shared__ float red[256];
  const int g = blockIdx.x;
  const int t = threadIdx.x;
  float s = 0.0f;
  for (int i = t; i < N_ROWS; i += 256) {
    int base = (i < B_ROWS) ? i : i - B_ROWS;
    if (gid[base] == g) s += sqd[i];
  }
  red[t] = s;
  __syncthreads();
  for (int off = 128; off; off >>= 1) {
    if (t < off) red[t] += red[t + off];
    __syncthreads();
  }
  if (t == 0) {
    float cnt = counts[g];
    float ms = red[0] / fmaxf(cnt, 1.0f);
    float d = ms / logf(cnt + 10.0f);
    density[g] = (cnt > 1.0f) ? d : 0.0f;
  }
}

// ---------------- K4: finalize density (max-fill, percentile clip, scale) ----------------
__global__ void density_finalize_kernel(const float* __restrict__ counts,
                                        float* __restrict__ density) {
  __shared__ float srt[K_GRP];
  __shared__ float red[K_GRP];
  const int t = threadIdx.x;
  float d = density[t];
  // max over entries (zeros where cnt<=1, density >= 0 so max unaffected)
  red[t] = d;
  __syncthreads();
  for (int off = 512; off; off >>= 1) {
    if (t < off) red[t] = fmaxf(red[t], red[t + off]);
    __syncthreads();
  }
  const float mx = red[0];
  __syncthreads();
  d = (counts[t] > 1.0f) ? d : mx;
  // bitonic sort a copy for exact percentiles
  srt[t] = d;
  __syncthreads();
  for (int k = 2; k <= K_GRP; k <<= 1) {
    for (int j = k >> 1; j > 0; j >>= 1) {
      int ix = t ^ j;
      if (ix > t) {
        bool asc = ((t & k) == 0);
        float a = srt[t], b = srt[ix];
        if ((a > b) == asc) { srt[t] = b; srt[ix] = a; }
      }
      __syncthreads();
    }
  }
  // linear-interp percentiles: pos = q/100*(n-1); n=1024
  const float p10 = srt[102] + 0.3f * (srt[103] - srt[102]); // pos 102.3
  const float p90 = srt[920] + 0.7f * (srt[921] - srt[920]); // pos 920.7
  float dc = fminf(fmaxf(d, p10), p90);
  red[t] = dc;
  __syncthreads();
  for (int off = 512; off; off >>= 1) {
    if (t < off) red[t] += red[t + off];
    __syncthreads();
  }
  const float mean = red[0] * (1.0f / (float)K_GRP);
  density[t] = 0.1f * dc / mean;
}

// ---------------- K5: fused f32-WMMA GEMM + masked log-softmax loss ----------------
// Block: 512 threads (16 waves), 32 rows x 1024 cols, D=512 reduction in steps of 4.
// Two 16-row A tiles share every B load -> 2 WMMAs per B fetch.
__global__ void gemm_loss_kernel(const float* __restrict__ X,
                                 const int* __restrict__ subject,
                                 const int* __restrict__ labels,
                                 const float* __restrict__ centroidT,
                                 const float* __restrict__ density,
                                 float* __restrict__ row_loss) {
  extern __shared__ float smem[];
  float* As = smem;                                   // 32 x APITCH A tiles
  float* Ss = smem + MROWS * APITCH;                  // 32 x SPITCH scaled sims
  float* Dd = smem + MROWS * APITCH + MROWS * SPITCH; // 1024 densities

  const int tid  = threadIdx.x;
  const int wave = tid >> 5;
  const int lane = tid & 31;
  const int half = lane >> 4;   // 0: lanes 0-15, 1: lanes 16-31
  const int ln   = lane & 15;
  const int m0   = blockIdx.x * MROWS;

  // stage densities
  for (int g = tid; g < K_GRP; g += 512) Dd[g] = density[g];

  // stage A tiles: row r = tid>>4 (0..31), 32 contiguous floats per thread
  {
    const int r = tid >> 4;
    const int ds = (tid & 15) * 32;
    const int R = m0 + r;
    const float* src = (R < B_ROWS) ? (X + (size_t)R * 1024 + ds)
                                    : (X + (size_t)(R - B_ROWS) * 1024 + 512 + ds);
    float* dst = As + r * APITCH + ds;
#pragma unroll
    for (int k = 0; k < 32; k += 4) {
      float4 v = *(const float4*)(src + k);
      *(float4*)(dst + k) = v;
    }
  }
  __syncthreads();

  // GEMM: each wave owns 4 column tiles (tile ids wave, wave+16, wave+32, wave+48)
  const float* a0row = As + ln * APITCH;        // rows 0-15,  M = lane%16
  const float* a1row = As + (16 + ln) * APITCH; // rows 16-31
#pragma unroll
  for (int tix = 0; tix < 4; ++tix) {
    const int nt   = wave + tix * 16;
    const int col0 = nt * 16;
    const float* bcol = centroidT + col0 + ln;  // column N = col0 + lane%16

    // prefetch part of the next tile's B columns (gfx1250 global_prefetch_b8)
    if (tix < 3) {
      const float* pnext = centroidT + (col0 + 256) + ln;
#pragma unroll
      for (int pk = 0; pk < 4; ++pk)
        __builtin_prefetch(pnext + (size_t)(lane + 128 * pk) * K_GRP, 0, 1);
    }

    v8f acc0 = {0.f, 0.f, 0.f, 0.f, 0.f, 0.f, 0.f, 0.f};
    v8f acc1 = {0.f, 0.f, 0.f, 0.f, 0.f, 0.f, 0.f, 0.f};
#pragma unroll 8
    for (int db = 0; db < DIMS; db += 4) {
      const int k0 = db + 2 * half; // lanes 0-15 take K0,K1; lanes 16-31 take K2,K3
      // A 16x4 f32 operands from LDS (one float2 per lane)
      v2f a0 = *(const v2f*)(a0row + k0);
      v2f a1 = *(const v2f*)(a1row + k0);
      // B 4x16 f32 operand from centroid^T: two coalesced dword loads per lane
      v2f b;
      b.x = bcol[(size_t)k0 * K_GRP];
      b.y = bcol[(size_t)(k0 + 1) * K_GRP];
      acc0 = __builtin_amdgcn_wmma_f32_16x16x4_f32(
          false, a0, false, b, (short)0, acc0, false, false);
      acc1 = __builtin_amdgcn_wmma_f32_16x16x4_f32(
          false, a1, false, b, (short)0, acc1, false, false);
    }
    // C layout: VGPR j, lanes 0-15 -> M=j, lanes 16-31 -> M=j+8; N = lane%16
    const float dv = Dd[col0 + ln];
#pragma unroll
    for (int j = 0; j < 8; ++j) {
      const int m = j + 8 * half;
      Ss[m * SPITCH + col0 + ln]        = acc0[j] / dv;
      Ss[(m + 16) * SPITCH + col0 + ln] = acc1[j] / dv;
    }
  }
  __syncthreads();

  // Per-row masked softmax loss: wave w handles rows 2w, 2w+1 (32 rows, 16 waves)
  for (int rr = 0; rr < 2; ++rr) {
    const int m = wave * 2 + rr;
    const int R = m0 + m;
    const int base = (R < B_ROWS) ? R : R - B_ROWS;
    const int s_row = subject[base];
    const int l_row = labels[base];
    const float* row = Ss + m * SPITCH;

    float mxv = -3.402823466e+38f;
    for (int c = lane; c < K_GRP; c += 32) mxv = fmaxf(mxv, row[c]);
    for (int off = 16; off; off >>= 1) mxv = fmaxf(mxv, __shfl_xor(mxv, off, 32));

    float se = 0.0f, sz = 0.0f, cm = 0.0f;
    for (int c = lane; c < K_GRP; c += 32) {
      const int gs = c >> 4;   // group subject
      const int gl = c & 15;   // group label
      const bool mf = (gs != s_row) && (gl == l_row);
      const float z = mf ? (row[c] - mxv) : 0.0f; // masked-out entries contribute exp(0)=1
      se += expf(z);
      sz += mf ? z : 0.0f;
      cm += mf ? 1.0f : 0.0f;
    }
    for (int off = 16; off; off >>= 1) {
      se += __shfl_xor(se, off, 32);
      sz += __shfl_xor(sz, off, 32);
      cm += __shfl_xor(cm, off, 32);
    }
    if (lane == 0) row_loss[R] = logf(se) - sz / cm;
  }
}

// ---------------- K6: mean of row losses ----------------
__global__ void final_mean_kernel(const float* __restrict__ row_loss,
                                  float* __restrict__ out) {
  __shared__ float red[256];
  const int t = threadIdx.x;
  float s = 0.0f;
  for (int i = t; i < N_ROWS; i += 256) s += row_loss[i];
  red[t] = s;
  __syncthreads();
  for (int off = 128; off; off >>= 1) {
    if (t < off) red[t] += red[t + off];
    __syncthreads();
  }
  if (t == 0) out[0] = red[0] * (1.0f / (float)N_ROWS);
}

extern "C" void kernel_launch(void* const* d_in, const int* in_sizes, int n_in,
                              void* d_out, int out_size, void* d_ws, size_t ws_size,
                              hipStream_t stream) {
  const float* X      = (const float*)d_in[0]; // [16384, 2, 512]
  const int* subject  = (const int*)d_in[1];   // [16384]
  const int* labels   = (const int*)d_in[2];   // [16384]
  float* out          = (float*)d_out;

  // workspace layout
  char* ws = (char*)d_ws;
  int*   gid       = (int*)(ws);                         // 64 KB
  float* counts    = (float*)(ws + 65536);               // 4 KB
  float* density   = (float*)(ws + 69632);               // 4 KB
  float* centroid  = (float*)(ws + 73728);               // 2 MB  [K][D]
  float* centroidT = (float*)(ws + 2170880);             // 2 MB  [D][K]
  float* sqd       = (float*)(ws + 4268032);             // 128 KB
  float* row_loss  = (float*)(ws + 4399104);             // 128 KB

  gid_kernel<<<(B_ROWS + 255) / 256, 256, 0, stream>>>(subject, labels, gid);
  centroid_kernel<<<K_GRP, 256, 0, stream>>>(X, gid, centroid, centroidT, counts);
  dist_kernel<<<N_ROWS / 8, 256, 0, stream>>>(X, gid, centroid, sqd);
  group_density_kernel<<<K_GRP, 256, 0, stream>>>(gid, sqd, counts, density);
  density_finalize_kernel<<<1, K_GRP, 0, stream>>>(counts, density);

  const size_t smem_bytes =
      (size_t)(MROWS * APITCH + MROWS * SPITCH + K_GRP) * sizeof(float);
  gemm_loss_kernel<<<N_ROWS / MROWS, 512, smem_bytes, stream>>>(
      X, subject, labels, centroidT, density, row_loss);

  final_mean_kernel<<<1, 256, 0, stream>>>(row_loss, out);
}